// MultiSlicePtychographyModel_5059471475185
// MI455X (gfx1250) — compile-verified
//
#include <hip/hip_runtime.h>

// Multi-slice ptychography forward model for MI455X (gfx1250).
// FFTs realized as DFT matmuls on the wave32 WMMA pipe (v_wmma_f32_16x16x4_f32).
// - wavefields resident in LDS (ping-pong complex buffers, stride 138 == 10 mod 64
//   -> conflict-free A/B fragment access and D-tile stores)
// - DFT matrices pre-swizzled into fragment order -> one coalesced b64 per F-frag
// - left multiply: wave owns a row strip (F A-frags hoisted over 8 col tiles)
//   right multiply: wave owns a col strip (F B-frags hoisted over 8 row tiles)
// - next-slice V patch prefetched (global_prefetch_b8) during transmit

#define MDIM   128
#define LSTR   138
#define NZC    8
#define NYC    512
#define NXC    512
#define KC     256
#define NMODESC 4

typedef float v2f __attribute__((ext_vector_type(2)));
typedef float v8f __attribute__((ext_vector_type(8)));

__device__ __forceinline__ v8f wmma_f32(v2f a, v2f b, v8f c) {
  // D = A(16x4) * B(4x16) + C, f32, wave32
  return __builtin_amdgcn_wmma_f32_16x16x4_f32(
      /*neg_a=*/false, a, /*neg_b=*/false, b,
      /*c_mod=*/(short)0, c, /*reuse_a=*/false, /*reuse_b=*/false);
}

// Pre-swizzled F fragment: layout [tile][kstep][lane] of v2f (coalesced 256B/wave)
__device__ __forceinline__ v2f load_ffrag(const float* __restrict__ p, int tile,
                                          int ks, int lane) {
  const v2f* q = (const v2f*)p;
  return q[(tile * 32 + ks) * 32 + lane];
}

// A-fragment (16x4, 32-bit) from LDS: lanes 0-15 K={k0,k0+1}, lanes 16-31 K={k0+2,k0+3}
__device__ __forceinline__ v2f load_afrag_lds(const float* __restrict__ p,
                                              int row0, int k0, int lane) {
  int half = lane >> 4, r = lane & 15;
  const float* q = p + (row0 + r) * LSTR + k0 + 2 * half;
  v2f a; a.x = q[0]; a.y = q[1];
  return a;
}

// B-fragment (4x16, 32-bit) from LDS
__device__ __forceinline__ v2f load_bfrag_lds(const float* __restrict__ p,
                                              int k0, int col0, int lane) {
  int half = lane >> 4, c = lane & 15;
  int kb = k0 + 2 * half;
  v2f b; b.x = p[kb * LSTR + col0 + c]; b.y = p[(kb + 1) * LSTR + col0 + c];
  return b;
}

__device__ __forceinline__ void store_tile(float* __restrict__ d, int row0, int col0,
                                           int lane, v8f acc) {
  int half = lane >> 4, c = lane & 15;
#pragma unroll
  for (int e = 0; e < 8; ++e) {
    int row = row0 + e + 8 * half;
    d[row * LSTR + col0 + c] = acc[e];
  }
}

// Left complex matmul: Y = Fc * X.  Yr = Fr@Xr + Fa@Xi ; Yi = Fr@Xi + Fb@Xr
// (forward: Fa=+sin, Fb=-sin ; inverse: Fa=-sin, Fb=+sin)
// Wave owns rows [16w,16w+16); F A-fragments hoisted across the 8 column tiles.
__device__ __forceinline__ void cmm_left(float* __restrict__ dr, float* __restrict__ di,
                                         const float* __restrict__ sr, const float* __restrict__ si,
                                         const float* __restrict__ FAr, const float* __restrict__ FAa,
                                         const float* __restrict__ FAb, int tid) {
  int lane = tid & 31;
  int wv = tid >> 5;
  int row0 = wv * 16;
  v8f accr[8], acci[8];
#pragma unroll
  for (int ct = 0; ct < 8; ++ct) {
    accr[ct] = (v8f){0.f, 0.f, 0.f, 0.f, 0.f, 0.f, 0.f, 0.f};
    acci[ct] = (v8f){0.f, 0.f, 0.f, 0.f, 0.f, 0.f, 0.f, 0.f};
  }
  for (int ks = 0; ks < 32; ++ks) {
    int k0 = ks * 4;
    v2f fr = load_ffrag(FAr, wv, ks, lane);
    v2f fa = load_ffrag(FAa, wv, ks, lane);
    v2f fb = load_ffrag(FAb, wv, ks, lane);
#pragma unroll
    for (int ct = 0; ct < 8; ++ct) {
      int col0 = ct * 16;
      v2f xr = load_bfrag_lds(sr, k0, col0, lane);
      v2f xi = load_bfrag_lds(si, k0, col0, lane);
      accr[ct] = wmma_f32(fr, xr, accr[ct]);
      acci[ct] = wmma_f32(fr, xi, acci[ct]);
      accr[ct] = wmma_f32(fa, xi, accr[ct]);
      acci[ct] = wmma_f32(fb, xr, acci[ct]);
    }
  }
#pragma unroll
  for (int ct = 0; ct < 8; ++ct) {
    store_tile(dr, row0, ct * 16, lane, accr[ct]);
    store_tile(di, row0, ct * 16, lane, acci[ct]);
  }
}

// Right complex matmul: Y = X * Fc.  Yr = Xr@Fr + Xi@Fa ; Yi = Xi@Fr + Xr@Fb
// Wave owns cols [16w,16w+16); F B-fragments hoisted across the 8 row tiles.
__device__ __forceinline__ void cmm_right(float* __restrict__ dr, float* __restrict__ di,
                                          const float* __restrict__ sr, const float* __restrict__ si,
                                          const float* __restrict__ FBr, const float* __restrict__ FBa,
                                          const float* __restrict__ FBb, int tid) {
  int lane = tid & 31;
  int wv = tid >> 5;
  int col0 = wv * 16;
  v8f accr[8], acci[8];
#pragma unroll
  for (int rt = 0; rt < 8; ++rt) {
    accr[rt] = (v8f){0.f, 0.f, 0.f, 0.f, 0.f, 0.f, 0.f, 0.f};
    acci[rt] = (v8f){0.f, 0.f, 0.f, 0.f, 0.f, 0.f, 0.f, 0.f};
  }
  for (int ks = 0; ks < 32; ++ks) {
    int k0 = ks * 4;
    v2f fr = load_ffrag(FBr, wv, ks, lane);
    v2f fa = load_ffrag(FBa, wv, ks, lane);
    v2f fb = load_ffrag(FBb, wv, ks, lane);
#pragma unroll
    for (int rt = 0; rt < 8; ++rt) {
      int row0 = rt * 16;
      v2f xr = load_afrag_lds(sr, row0, k0, lane);
      v2f xi = load_afrag_lds(si, row0, k0, lane);
      accr[rt] = wmma_f32(xr, fr, accr[rt]);
      acci[rt] = wmma_f32(xi, fr, acci[rt]);
      accr[rt] = wmma_f32(xi, fa, accr[rt]);
      acci[rt] = wmma_f32(xr, fb, acci[rt]);
    }
  }
#pragma unroll
  for (int rt = 0; rt < 8; ++rt) {
    store_tile(dr, rt * 16, col0, lane, accr[rt]);
    store_tile(di, rt * 16, col0, lane, acci[rt]);
  }
}

// exact twiddle: W^{rc} periodic mod 128
__device__ __forceinline__ void fval(int r, int c, float* co, float* si) {
  int t = (r * c) & 127;
  float ang = (float)t * (6.28318530717958647692f / 128.0f);
  __sincosf(ang, si, co);
}

// Build fragment-swizzled DFT tables (cos, +sin, -sin) for A-side and B-side use,
// plus the pre-scaled Fresnel propagator H/N^2.
extern "C" __global__ void ptycho_init(float* __restrict__ FAr, float* __restrict__ FAs,
                                       float* __restrict__ FAn, float* __restrict__ FBr,
                                       float* __restrict__ FBs, float* __restrict__ FBn,
                                       float* __restrict__ Hr, float* __restrict__ Hi) {
  int idx = blockIdx.x * blockDim.x + threadIdx.x;
  if (idx >= MDIM * MDIM) return;

  if (idx < 8192) {  // one v2f fragment slot: [tile][kstep][lane]
    int tile = idx >> 10;
    int ks = (idx >> 5) & 31;
    int lane = idx & 31;
    int half = lane >> 4, r = lane & 15;

    // A-side fragment: rows tile*16+r, K = ks*4 + 2*half (+0,+1)
    int arow = tile * 16 + r;
    int ak = ks * 4 + 2 * half;
    float c0, s0, c1, s1;
    fval(arow, ak, &c0, &s0);
    fval(arow, ak + 1, &c1, &s1);
    FAr[2 * idx] = c0;  FAr[2 * idx + 1] = c1;
    FAs[2 * idx] = s0;  FAs[2 * idx + 1] = s1;
    FAn[2 * idx] = -s0; FAn[2 * idx + 1] = -s1;

    // B-side fragment: rows K = ks*4 + 2*half (+0,+1), col tile*16+r
    int bk = ks * 4 + 2 * half;
    int bcol = tile * 16 + r;
    fval(bk, bcol, &c0, &s0);
    fval(bk + 1, bcol, &c1, &s1);
    FBr[2 * idx] = c0;  FBr[2 * idx + 1] = c1;
    FBs[2 * idx] = s0;  FBs[2 * idx + 1] = s1;
    FBn[2 * idx] = -s0; FBn[2 * idx + 1] = -s1;
  }

  // propagator on fftfreq grid, with inverse-fft 1/N^2 folded in
  int j = idx >> 7, x = idx & 127;
  int fj = (j < 64) ? j : j - 128;
  int fx = (x < 64) ? x : x - 128;
  const float dfi = 1.0f / (128.0f * 0.2f);  // 1/(M*DR)
  float fy1 = (float)fj * dfi, fx1 = (float)fx * dfi;
  float f2 = fy1 * fy1 + fx1 * fx1;
  float phase = -3.14159265358979323846f * 0.025f * 10.0f * f2;  // -pi*lambda*dz*f2
  float hs, hc;
  __sincosf(phase, &hs, &hc);
  const float invN2 = 1.0f / 16384.0f;
  Hr[idx] = hc * invN2;
  Hi[idx] = hs * invN2;
}

extern "C" __global__ void __launch_bounds__(256)
ptycho_main(const float* __restrict__ V, const float* __restrict__ pre,
            const float* __restrict__ pim, const int* __restrict__ pos,
            const float* __restrict__ FAr, const float* __restrict__ FAs,
            const float* __restrict__ FAn, const float* __restrict__ FBr,
            const float* __restrict__ FBs, const float* __restrict__ FBn,
            const float* __restrict__ Hr, const float* __restrict__ Hi,
            float* __restrict__ out) {
  extern __shared__ float lds[];             // 4 * 128*138 floats = 282,624 B (< 320 KB/WGP)
  float* Ar = lds;
  float* Ai = lds + MDIM * LSTR;
  float* Br = lds + 2 * MDIM * LSTR;
  float* Bi = lds + 3 * MDIM * LSTR;
  const int tid = threadIdx.x;
  const int k = blockIdx.x;
  const int p0 = pos[2 * k], p1 = pos[2 * k + 1];

  for (int mode = 0; mode < NMODESC; ++mode) {
    // load probe mode into wavefield buffers
    for (int j = 0; j < 64; ++j) {
      int idx = tid + 256 * j;
      int y = idx >> 7, x = idx & 127;
      Ar[y * LSTR + x] = pre[mode * MDIM * MDIM + idx];
      Ai[y * LSTR + x] = pim[mode * MDIM * MDIM + idx];
    }
    __syncthreads();

    for (int z = 0; z < NZC - 1; ++z) {
      // transmit: w *= exp(i * V[z, p0+y, p1+x]); prefetch next slice's patch
      for (int j = 0; j < 64; ++j) {
        int idx = tid + 256 * j;
        int y = idx >> 7, x = idx & 127;
        size_t voff = (size_t)z * NYC * NXC + (size_t)(p0 + y) * NXC + (p1 + x);
        float v = V[voff];
        __builtin_prefetch(&V[voff + (size_t)NYC * NXC], 0, 3);  // global_prefetch_b8
        float sv, cv;
        __sincosf(v, &sv, &cv);
        int a = y * LSTR + x;
        float wr = Ar[a], wi = Ai[a];
        Ar[a] = wr * cv - wi * sv;
        Ai[a] = wr * sv + wi * cv;
      }
      __syncthreads();
      // forward fft2 = F * X * F
      cmm_left (Br, Bi, Ar, Ai, FAr, FAs, FAn, tid); __syncthreads();
      cmm_right(Ar, Ai, Br, Bi, FBr, FBs, FBn, tid); __syncthreads();
      // multiply by propagator (includes 1/N^2 of the inverse transform)
      for (int j = 0; j < 64; ++j) {
        int idx = tid + 256 * j;
        int y = idx >> 7, x = idx & 127;
        float hr = Hr[idx], hi = Hi[idx];
        int a = y * LSTR + x;
        float wr = Ar[a], wi = Ai[a];
        Ar[a] = wr * hr - wi * hi;
        Ai[a] = wr * hi + wi * hr;
      }
      __syncthreads();
      // inverse fft2 (unscaled conj transform): swap +sin / -sin roles
      cmm_left (Br, Bi, Ar, Ai, FAr, FAn, FAs, tid); __syncthreads();
      cmm_right(Ar, Ai, Br, Bi, FBr, FBn, FBs, tid); __syncthreads();
    }

    // last slice: transmit only
    {
      const int z = NZC - 1;
      for (int j = 0; j < 64; ++j) {
        int idx = tid + 256 * j;
        int y = idx >> 7, x = idx & 127;
        float v = V[(size_t)z * NYC * NXC + (size_t)(p0 + y) * NXC + (p1 + x)];
        float sv, cv;
        __sincosf(v, &sv, &cv);
        int a = y * LSTR + x;
        float wr = Ar[a], wi = Ai[a];
        Ar[a] = wr * cv - wi * sv;
        Ai[a] = wr * sv + wi * cv;
      }
      __syncthreads();
    }

    // final forward fft2 (ortho norm folded into intensity scale below)
    cmm_left (Br, Bi, Ar, Ai, FAr, FAs, FAn, tid); __syncthreads();
    cmm_right(Ar, Ai, Br, Bi, FBr, FBs, FBn, tid); __syncthreads();

    // incoherent mode accumulation; same thread owns each pixel -> no atomics
    const float invO = 1.0f / 16384.0f;      // ortho: amp/128 -> intensity/16384
    for (int j = 0; j < 64; ++j) {
      int idx = tid + 256 * j;
      int y = idx >> 7, x = idx & 127;
      int a = y * LSTR + x;
      float wr = Ar[a], wi = Ai[a];
      float s = (wr * wr + wi * wi) * invO;
      size_t o = (size_t)k * MDIM * MDIM + idx;
      if (mode == 0) {
        out[o] = s;
      } else if (mode < NMODESC - 1) {
        out[o] += s;
      } else {
        out[o] = sqrtf(1e-10f + out[o] + s);
      }
    }
    __syncthreads();   // buffers reused by next mode
  }
}

extern "C" void kernel_launch(void* const* d_in, const int* in_sizes, int n_in,
                              void* d_out, int out_size, void* d_ws, size_t ws_size,
                              hipStream_t stream) {
  const float* V   = (const float*)d_in[0];
  const float* pre = (const float*)d_in[1];
  const float* pim = (const float*)d_in[2];
  const int*   pos = (const int*)d_in[3];

  float* ws = (float*)d_ws;
  float* FAr = ws;
  float* FAs = ws + 16384;
  float* FAn = ws + 2 * 16384;
  float* FBr = ws + 3 * 16384;
  float* FBs = ws + 4 * 16384;
  float* FBn = ws + 5 * 16384;
  float* Hr  = ws + 6 * 16384;
  float* Hi  = ws + 7 * 16384;

  ptycho_init<<<64, 256, 0, stream>>>(FAr, FAs, FAn, FBr, FBs, FBn, Hr, Hi);

  const size_t lds_bytes = 4u * MDIM * LSTR * sizeof(float);  // 282,624 B
  ptycho_main<<<KC, 256, lds_bytes, stream>>>(V, pre, pim, pos, FAr, FAs, FAn,
                                              FBr, FBs, FBn, Hr, Hi, (float*)d_out);
}